// DKTModel_3212635538065
// MI455X (gfx1250) — compile-verified
//
#include <hip/hip_runtime.h>
#include <math.h>

// ---------------------------------------------------------------------------
// DKT model: one-hot encode -> LSTM layer 0 -> LSTM layer 1 -> FC -> sigmoid
// fp32 WMMA (v_wmma_f32_16x16x4_f32) everywhere. Persistent LSTM kernel with
// per-timestep grid barrier. Weights staged in LDS (320KB/WGP on CDNA5) with
// bank-conflict-free padding; one-hot input folded into gate epilogue.
// ---------------------------------------------------------------------------

typedef __attribute__((ext_vector_type(2))) float v2f;
typedef __attribute__((ext_vector_type(8))) float v8f;

#define N_SKILLS_  1024
#define HIDDEN_    512
#define BATCH_     64
#define SEQ_       512
#define INPUT_DIM_ 2048   // 2 * N_SKILLS
#define WSTRIDE_   516    // 512 + 4 pad floats: b64 frag reads hit distinct banks

__device__ __forceinline__ v8f wmma_f32(v2f a, v2f b, v8f c) {
  // (neg_a, A, neg_b, B, c_mod, C, reuse_a, reuse_b)
  return __builtin_amdgcn_wmma_f32_16x16x4_f32(false, a, false, b, (short)0, c,
                                               false, false);
}

__device__ __forceinline__ float sigmoidf_(float x) {
  return 1.0f / (1.0f + expf(-x));
}

// Device-wide barrier: per-timestep counter, zeroed once per kernel_launch.
__device__ __forceinline__ void grid_barrier(int* bar, int nwg) {
  __threadfence();
  __syncthreads();
  if (threadIdx.x == 0) {
    __hip_atomic_fetch_add(bar, 1, __ATOMIC_ACQ_REL, __HIP_MEMORY_SCOPE_AGENT);
    while (__hip_atomic_load(bar, __ATOMIC_ACQUIRE, __HIP_MEMORY_SCOPE_AGENT) < nwg) {
      __builtin_amdgcn_s_sleep(2);
    }
  }
  __syncthreads();
  __threadfence();
}

__global__ void init_ws_kernel(int* bar, int n) {
  int i = blockIdx.x * blockDim.x + threadIdx.x;
  if (i < n) bar[i] = 0;
}

// ---------------------------------------------------------------------------
// Persistent LSTM layer. Grid = HIDDEN/16 = 32 workgroups, 128 threads each
// (wave w handles gate type w: i,f,g,o). Each wave accumulates the full 64x16
// gate tile (4 M-tiles of 16x16) over K=HIDDEN with fp32 WMMA. Weight tiles
// live in LDS for the whole sequence.
// ---------------------------------------------------------------------------
__global__ __launch_bounds__(128)
void lstm_layer_kernel(const float* __restrict__ Wih,   // [4H, IN] (l0) / [4H, H] (l1)
                       const float* __restrict__ Whh,   // [4H, H]
                       const float* __restrict__ b_ih,  // [4H]
                       const float* __restrict__ b_hh,  // [4H]
                       const int*   __restrict__ skills,   // [B, S]
                       const int*   __restrict__ corrects, // [B, S]
                       const float* __restrict__ x_seq,    // [S, B, H] (layer-1 input)
                       float*       __restrict__ h_seq,    // [S, B, H] output
                       float*       __restrict__ c_state,  // [B, H]
                       int*         __restrict__ bar,      // [S]
                       int layer) {
  extern __shared__ char smem_raw[];
  float* w_hh_s = (float*)smem_raw;                 // [64][WSTRIDE_]
  float* w_ih_s = w_hh_s + 64 * WSTRIDE_;           // [64][WSTRIDE_] (layer 1 only)
  float* lds_g  = (layer != 0) ? (w_ih_s + 64 * WSTRIDE_) : w_ih_s;  // [4][64][16]
  int*   idx_s  = (int*)(lds_g + 4 * BATCH_ * 16);  // [64]

  const int hc0  = blockIdx.x * 16;        // hidden-column tile base
  const int wave = threadIdx.x >> 5;       // gate type 0..3
  const int lane = threadIdx.x & 31;
  const int col  = lane & 15;              // N index within tile / M row for A
  const int hi   = lane >> 4;              // lane half: K pair for A/B, M+8 for C
  const int koff = hi * 2;
  const int n0   = wave * HIDDEN_ + hc0;   // gate column base in [0, 4H)
  const int nwg  = (int)gridDim.x;

  // ---- Stage this block's weight rows into LDS (once for all 512 steps) ----
  // Local row lr = gate*16 + c  ->  global row gr = gate*H + hc0 + c.
  for (int i = threadIdx.x; i < 64 * (HIDDEN_ / 4); i += blockDim.x) {
    int lr = i / (HIDDEN_ / 4);
    int k4 = (i % (HIDDEN_ / 4)) * 4;
    int gr = (lr >> 4) * HIDDEN_ + hc0 + (lr & 15);
    *(float4*)(&w_hh_s[lr * WSTRIDE_ + k4]) =
        *(const float4*)(Whh + (size_t)gr * HIDDEN_ + k4);
  }
  if (layer != 0) {
    for (int i = threadIdx.x; i < 64 * (HIDDEN_ / 4); i += blockDim.x) {
      int lr = i / (HIDDEN_ / 4);
      int k4 = (i % (HIDDEN_ / 4)) * 4;
      int gr = (lr >> 4) * HIDDEN_ + hc0 + (lr & 15);
      *(float4*)(&w_ih_s[lr * WSTRIDE_ + k4]) =
          *(const float4*)(Wih + (size_t)gr * HIDDEN_ + k4);
    }
  }
  __syncthreads();

  const float* bw_hh = &w_hh_s[(wave * 16 + col) * WSTRIDE_];
  const float* bw_ih = &w_ih_s[(wave * 16 + col) * WSTRIDE_];

  // Gate bias is time-invariant: hoist out of the sequence loop.
  const int gidx = n0 + col;
  const float bias = b_ih[gidx] + b_hh[gidx];

  const v8f vzero = {};

  for (int t = 0; t < SEQ_; ++t) {
    // Stage one-hot indices for this timestep (layer 0 only).
    if (layer == 0 && threadIdx.x < BATCH_) {
      int b  = threadIdx.x;
      int sk = skills[b * SEQ_ + t];
      int cr = corrects[b * SEQ_ + t];
      idx_s[b] = (sk >= 0) ? (sk + N_SKILLS_ * (1 - (cr == 1 ? 1 : 0))) : -1;
    }

    v8f acc[4];
#pragma unroll
    for (int mt = 0; mt < 4; ++mt) acc[mt] = vzero;

    // Recurrent term: h_{t-1} @ Whh^T  (skipped at t==0, h0 == 0)
    if (t > 0) {
      const float* hp = h_seq + (size_t)(t - 1) * BATCH_ * HIDDEN_;
      for (int k0 = 0; k0 < HIDDEN_; k0 += 4) {
        v2f bfrag = *(const v2f*)(bw_hh + k0 + koff);   // ds_load_b64
#pragma unroll
        for (int mt = 0; mt < 4; ++mt) {
          const float* arow = hp + (size_t)(mt * 16 + col) * HIDDEN_;
          v2f afrag = *(const v2f*)(arow + k0 + koff);  // global_load_b64 (L2-hot)
          acc[mt] = wmma_f32(afrag, bfrag, acc[mt]);
        }
      }
    }

    // Layer-1 input term: x_t @ Wih^T (dense GEMM from LDS weights)
    if (layer != 0) {
      const float* xp = x_seq + (size_t)t * BATCH_ * HIDDEN_;
      for (int k0 = 0; k0 < HIDDEN_; k0 += 4) {
        v2f bfrag = *(const v2f*)(bw_ih + k0 + koff);
#pragma unroll
        for (int mt = 0; mt < 4; ++mt) {
          const float* arow = xp + (size_t)(mt * 16 + col) * HIDDEN_;
          v2f afrag = *(const v2f*)(arow + k0 + koff);
          acc[mt] = wmma_f32(afrag, bfrag, acc[mt]);
        }
      }
    }

    __syncthreads();   // idx_s visible before epilogue reads it

    // Epilogue: bias (+ one-hot gather for layer 0), stage preacts to LDS.
#pragma unroll
    for (int mt = 0; mt < 4; ++mt) {
#pragma unroll
      for (int r = 0; r < 8; ++r) {
        int b = mt * 16 + r + 8 * hi;        // C layout: M = r + 8*(lane>>4)
        float pre = acc[mt][r] + bias;
        if (layer == 0) {
          int ix = idx_s[b];
          if (ix >= 0) pre += Wih[(size_t)gidx * INPUT_DIM_ + ix];
        }
        lds_g[(wave * BATCH_ + b) * 16 + col] = pre;
      }
    }
    __syncthreads();

    // Elementwise LSTM cell for this block's 64x16 (batch x hidden) slice.
    for (int e = threadIdx.x; e < BATCH_ * 16; e += blockDim.x) {
      int b  = e >> 4;
      int c  = e & 15;
      int hu = hc0 + c;
      float gi = lds_g[(0 * BATCH_ + b) * 16 + c];
      float gf = lds_g[(1 * BATCH_ + b) * 16 + c];
      float gg = lds_g[(2 * BATCH_ + b) * 16 + c];
      float go = lds_g[(3 * BATCH_ + b) * 16 + c];
      float si = sigmoidf_(gi);
      float sf = sigmoidf_(gf);
      float tg = tanhf(gg);
      float so = sigmoidf_(go);
      float cold = (t == 0) ? 0.0f : c_state[b * HIDDEN_ + hu];
      float cnew = sf * cold + si * tg;
      float hnew = so * tanhf(cnew);
      c_state[b * HIDDEN_ + hu] = cnew;
      h_seq[(size_t)t * BATCH_ * HIDDEN_ + (size_t)b * HIDDEN_ + hu] = hnew;
    }

    grid_barrier(bar + t, nwg);   // make h_t visible device-wide before t+1
  }
}

// ---------------------------------------------------------------------------
// Projection: logits = h2 @ fc_W^T + fc_b, sigmoid, emit in [B, S, N] order.
// One 16x16 tile per wave; rows of h2_seq are (s*B + b). Bandwidth-bound
// (~192 MB @ 23.3 TB/s), fp32 WMMA is more than enough compute.
// ---------------------------------------------------------------------------
__global__ __launch_bounds__(128)
void proj_kernel(const float* __restrict__ h2_seq,  // [S*B, H]
                 const float* __restrict__ fc_W,    // [N, H]
                 const float* __restrict__ fc_b,    // [N]
                 float*       __restrict__ out) {   // [B, S, N]
  const int wave = threadIdx.x >> 5;
  const int lane = threadIdx.x & 31;
  const int col  = lane & 15;
  const int hi   = lane >> 4;
  const int koff = hi * 2;

  const int ntiles_n = N_SKILLS_ / 16;               // 64
  const int tile = (int)blockIdx.x * 4 + wave;
  const int m0 = (tile / ntiles_n) * 16;             // row base in [0, S*B)
  const int n0 = (tile % ntiles_n) * 16;

  const float* wrow = fc_W   + (size_t)(n0 + col) * HIDDEN_;
  const float* arow = h2_seq + (size_t)(m0 + col) * HIDDEN_;

  v8f acc = {};
  for (int k0 = 0; k0 < HIDDEN_; k0 += 4) {
    v2f a = *(const v2f*)(arow + k0 + koff);
    v2f b = *(const v2f*)(wrow + k0 + koff);
    acc = wmma_f32(a, b, acc);
  }

#pragma unroll
  for (int r = 0; r < 8; ++r) {
    int row = m0 + r + 8 * hi;        // row = s*BATCH + b
    int s = row >> 6;
    int b = row & 63;
    int n = n0 + col;
    float logit = acc[r] + fc_b[n];
    out[(size_t)b * SEQ_ * N_SKILLS_ + (size_t)s * N_SKILLS_ + n] = sigmoidf_(logit);
  }
}

// ---------------------------------------------------------------------------
extern "C" void kernel_launch(void* const* d_in, const int* in_sizes, int n_in,
                              void* d_out, int out_size, void* d_ws, size_t ws_size,
                              hipStream_t stream) {
  (void)in_sizes; (void)n_in; (void)out_size; (void)ws_size;

  const int*   skills   = (const int*)d_in[0];
  const int*   corrects = (const int*)d_in[1];
  const float* W_ih0 = (const float*)d_in[2];
  const float* W_hh0 = (const float*)d_in[3];
  const float* b_ih0 = (const float*)d_in[4];
  const float* b_hh0 = (const float*)d_in[5];
  const float* W_ih1 = (const float*)d_in[6];
  const float* W_hh1 = (const float*)d_in[7];
  const float* b_ih1 = (const float*)d_in[8];
  const float* b_hh1 = (const float*)d_in[9];
  const float* fc_W  = (const float*)d_in[10];
  const float* fc_b  = (const float*)d_in[11];
  float* out = (float*)d_out;

  // Workspace carve-up (floats): h1_seq | h2_seq | c0 | c1 | barriers
  float* h1_seq = (float*)d_ws;                                  // S*B*H
  float* h2_seq = h1_seq + (size_t)SEQ_ * BATCH_ * HIDDEN_;      // S*B*H
  float* c0     = h2_seq + (size_t)SEQ_ * BATCH_ * HIDDEN_;      // B*H
  float* c1     = c0 + (size_t)BATCH_ * HIDDEN_;                 // B*H
  int*   bar0   = (int*)(c1 + (size_t)BATCH_ * HIDDEN_);         // S
  int*   bar1   = bar0 + SEQ_;                                   // S

  // Dynamic LDS: layer0 = Whh tile + gates + idx; layer1 adds Wih tile.
  const size_t smem_w   = (size_t)64 * WSTRIDE_ * sizeof(float);     // 132096
  const size_t smem_g   = (size_t)4 * BATCH_ * 16 * sizeof(float);   // 16384
  const size_t smem_idx = (size_t)BATCH_ * sizeof(int);              // 256
  const size_t smem_l0  = smem_w + smem_g + smem_idx;                // ~149 KB
  const size_t smem_l1  = 2 * smem_w + smem_g + smem_idx;            // ~281 KB (<=320KB)

  (void)hipFuncSetAttribute((const void*)lstm_layer_kernel,
                            hipFuncAttributeMaxDynamicSharedMemorySize,
                            (int)smem_l1);

  // Zero barrier counters every call (graph-replay safe).
  init_ws_kernel<<<(2 * SEQ_ + 255) / 256, 256, 0, stream>>>(bar0, 2 * SEQ_);

  // Layer 0: one-hot gather input + recurrent WMMA GEMM (persistent).
  lstm_layer_kernel<<<HIDDEN_ / 16, 128, smem_l0, stream>>>(
      W_ih0, W_hh0, b_ih0, b_hh0, skills, corrects, nullptr,
      h1_seq, c0, bar0, 0);

  // Layer 1: dense input GEMM + recurrent GEMM (persistent).
  lstm_layer_kernel<<<HIDDEN_ / 16, 128, smem_l1, stream>>>(
      W_ih1, W_hh1, b_ih1, b_hh1, skills, corrects, h1_seq,
      h2_seq, c1, bar1, 1);

  // Output projection + sigmoid.
  const int total_tiles = (SEQ_ * BATCH_ / 16) * (N_SKILLS_ / 16);  // 131072
  proj_kernel<<<total_tiles / 4, 128, 0, stream>>>(h2_seq, fc_W, fc_b, out);
}